// mlp_proj_layer_30502857736499
// MI455X (gfx1250) — compile-verified
//
#include <hip/hip_runtime.h>

typedef float v2f __attribute__((ext_vector_type(2)));
typedef float v4f __attribute__((ext_vector_type(4)));
typedef float v8f __attribute__((ext_vector_type(8)));

#define B_SZ 32
#define L_CH 96
#define NE   1024
#define MS   2048
#define NW   110592
#define KC   128
#define KCP  132      // padded LDS row stride (dwords): 16B-aligned rows, conflict-free b64 reads

__device__ __forceinline__ int chunk_size(int l) {
  return (l < 48) ? 1024 : (l < 72 ? 2048 : 512);
}
__device__ __forceinline__ int chunk_off(int l) {
  return (l < 48) ? (l << 10)
       : (l < 72) ? (49152 + ((l - 48) << 11))
                  : (98304 + ((l - 72) << 9));
}

// ---------------------------------------------------------------------------
// Encoder: per chunk l, reps[0:32, l, n0:n0+256] = x[0:32, off:off+S] @ Wenc[l]
// grid = (96, 4), block = 256 (8 wave32). Each wave owns 32 N-columns and the
// full M=32 (two 16x16 tiles) so every weight load feeds two WMMAs.
// ---------------------------------------------------------------------------
__global__ __launch_bounds__(256)
void enc_kernel(const float* __restrict__ params,
                const float* __restrict__ Wenc,
                float* __restrict__ reps)
{
  __shared__ float a_lds[B_SZ * KCP];

  const int l    = blockIdx.x;
  const int n0   = blockIdx.y * 256;
  const int S    = chunk_size(l);
  const int off  = chunk_off(l);
  const int tid  = threadIdx.x;
  const int wave = tid >> 5;
  const int lane = tid & 31;
  const int half = lane >> 4;        // 0: lanes 0-15, 1: lanes 16-31
  const int mrow = lane & 15;
  const int koff = half * 2;         // A/B fragment K offset per ISA layout

  const float* wbase = Wenc + (size_t)l * MS * NE;
  const int    ncol0 = n0 + wave * 32 + mrow;

  v8f acc[2][2] = {};                // [ntile][mtile]

  for (int kc = 0; kc < S; kc += KC) {
    // stage x tile [32 x 128] into LDS (fully coalesced b128 per wave-row)
    #pragma unroll
    for (int it = 0; it < 4; ++it) {
      int idx = tid + it * 256;
      int row = idx >> 5;
      int c4  = idx & 31;
      v4f v = *(const v4f*)(params + (size_t)row * NW + off + kc + c4 * 4);
      *(v4f*)&a_lds[row * KCP + c4 * 4] = v;
    }
    __syncthreads();

    #pragma unroll 4
    for (int k = 0; k < KC; k += 4) {
      // A fragments (16x4 f32): lane reads 2 consecutive K values of its row
      v2f afrag[2];
      #pragma unroll
      for (int mt = 0; mt < 2; ++mt)
        afrag[mt] = *(const v2f*)&a_lds[(mt * 16 + mrow) * KCP + k + koff];

      const int kb = kc + k + koff;
      #pragma unroll
      for (int nt = 0; nt < 2; ++nt) {
        // B fragment (4x16 f32): rows kb, kb+1 at this lane's column, streaming NT
        const float* bp = wbase + (size_t)kb * NE + (ncol0 + nt * 16);
        v2f bfrag;
        bfrag.x = __builtin_nontemporal_load(bp);
        bfrag.y = __builtin_nontemporal_load(bp + NE);
        #pragma unroll
        for (int mt = 0; mt < 2; ++mt)
          acc[nt][mt] = __builtin_amdgcn_wmma_f32_16x16x4_f32(
              false, afrag[mt], false, bfrag, (short)0, acc[nt][mt], false, false);
      }
    }
    __syncthreads();
  }

  // write reps[b][l][d] per 16x16 f32 C/D layout
  #pragma unroll
  for (int nt = 0; nt < 2; ++nt) {
    int d = n0 + wave * 32 + nt * 16 + mrow;
    #pragma unroll
    for (int mt = 0; mt < 2; ++mt) {
      #pragma unroll
      for (int r = 0; r < 8; ++r) {
        int m = mt * 16 + r + half * 8;
        reps[(size_t)m * (L_CH * NE) + (size_t)l * NE + d] = acc[nt][mt][r];
      }
    }
  }
}

// ---------------------------------------------------------------------------
// LayerNorm over 1024-dim rows of reps, in place. grid = 3072, block = 256.
// ---------------------------------------------------------------------------
__global__ __launch_bounds__(256)
void ln_kernel(float* __restrict__ reps,
               const float* __restrict__ gamma,
               const float* __restrict__ beta)
{
  __shared__ float red[16];
  float* p = reps + (size_t)blockIdx.x * NE;
  const int tid = threadIdx.x;

  float x[4];
  float s = 0.f, s2 = 0.f;
  #pragma unroll
  for (int i = 0; i < 4; ++i) {
    x[i] = p[tid + i * 256];
    s  += x[i];
    s2 += x[i] * x[i];
  }
  #pragma unroll
  for (int o = 16; o > 0; o >>= 1) {
    s  += __shfl_xor(s,  o, 32);
    s2 += __shfl_xor(s2, o, 32);
  }
  if ((tid & 31) == 0) { red[tid >> 5] = s; red[8 + (tid >> 5)] = s2; }
  __syncthreads();
  if (tid < 32) {
    float a = (tid < 8) ? red[tid]     : 0.f;
    float b = (tid < 8) ? red[8 + tid] : 0.f;
    #pragma unroll
    for (int o = 4; o > 0; o >>= 1) {
      a += __shfl_xor(a, o, 32);
      b += __shfl_xor(b, o, 32);
    }
    if (tid == 0) { red[0] = a; red[1] = b; }
  }
  __syncthreads();
  const float mu  = red[0] * (1.f / NE);
  const float var = red[1] * (1.f / NE) - mu * mu;
  const float inv = rsqrtf(var + 1e-5f);
  #pragma unroll
  for (int i = 0; i < 4; ++i) {
    int d = tid + i * 256;
    p[d] = (x[i] - mu) * inv * gamma[d] + beta[d];
  }
}

// ---------------------------------------------------------------------------
// Decoder: out[0:32, off+n0 : off+n0+256] = reps[0:32, l, :] @ Wdec[l][:, n0:]
// grid = (96, 8), block = 256. Tiles beyond the chunk size exit early.
// ---------------------------------------------------------------------------
__global__ __launch_bounds__(256)
void dec_kernel(const float* __restrict__ reps,
                const float* __restrict__ Wdec,
                float* __restrict__ out)
{
  __shared__ float a_lds[B_SZ * KCP];

  const int l  = blockIdx.x;
  const int S  = chunk_size(l);
  const int n0 = blockIdx.y * 256;
  if (n0 >= S) return;               // chunk sizes are multiples of 256
  const int off  = chunk_off(l);
  const int tid  = threadIdx.x;
  const int wave = tid >> 5;
  const int lane = tid & 31;
  const int half = lane >> 4;
  const int mrow = lane & 15;
  const int koff = half * 2;

  const float* wbase = Wdec + (size_t)l * NE * MS;
  const float* abase = reps + (size_t)l * NE;
  const int    ncol0 = n0 + wave * 32 + mrow;

  v8f acc[2][2] = {};

  for (int kc = 0; kc < NE; kc += KC) {
    #pragma unroll
    for (int it = 0; it < 4; ++it) {
      int idx = tid + it * 256;
      int row = idx >> 5;
      int c4  = idx & 31;
      v4f v = *(const v4f*)(abase + (size_t)row * (L_CH * NE) + kc + c4 * 4);
      *(v4f*)&a_lds[row * KCP + c4 * 4] = v;
    }
    __syncthreads();

    #pragma unroll 4
    for (int k = 0; k < KC; k += 4) {
      v2f afrag[2];
      #pragma unroll
      for (int mt = 0; mt < 2; ++mt)
        afrag[mt] = *(const v2f*)&a_lds[(mt * 16 + mrow) * KCP + k + koff];

      const int kb = kc + k + koff;
      #pragma unroll
      for (int nt = 0; nt < 2; ++nt) {
        const float* bp = wbase + (size_t)kb * MS + (ncol0 + nt * 16);
        v2f bfrag;
        bfrag.x = __builtin_nontemporal_load(bp);
        bfrag.y = __builtin_nontemporal_load(bp + MS);
        #pragma unroll
        for (int mt = 0; mt < 2; ++mt)
          acc[nt][mt] = __builtin_amdgcn_wmma_f32_16x16x4_f32(
              false, afrag[mt], false, bfrag, (short)0, acc[nt][mt], false, false);
      }
    }
    __syncthreads();
  }

  // scatter to flat output at this chunk's offset
  #pragma unroll
  for (int nt = 0; nt < 2; ++nt) {
    int n = n0 + wave * 32 + nt * 16 + mrow;
    #pragma unroll
    for (int mt = 0; mt < 2; ++mt) {
      #pragma unroll
      for (int r = 0; r < 8; ++r) {
        int m = mt * 16 + r + half * 8;
        out[(size_t)m * NW + off + n] = acc[nt][mt][r];
      }
    }
  }
}

extern "C" void kernel_launch(void* const* d_in, const int* in_sizes, int n_in,
                              void* d_out, int out_size, void* d_ws, size_t ws_size,
                              hipStream_t stream) {
  const float* params = (const float*)d_in[0];   // [32, 110592]
  const float* Wenc   = (const float*)d_in[1];   // [96, 2048, 1024]
  const float* Wdec   = (const float*)d_in[2];   // [96, 1024, 2048]
  const float* gamma  = (const float*)d_in[3];   // [1024]
  const float* beta   = (const float*)d_in[4];   // [1024]
  float*       out    = (float*)d_out;           // [32, 110592]
  float*       reps   = (float*)d_ws;            // [32, 96, 1024] = 12.6 MB scratch

  enc_kernel<<<dim3(L_CH, 4), 256, 0, stream>>>(params, Wenc, reps);
  ln_kernel<<<dim3(B_SZ * L_CH), 256, 0, stream>>>(reps, gamma, beta);
  dec_kernel<<<dim3(L_CH, 8), 256, 0, stream>>>(reps, Wdec, out);
}